// SparseLRRLayer_laplace_86088324481669
// MI455X (gfx1250) — compile-verified
//
#include <hip/hip_runtime.h>

#define N_NODES 4096
#define D_DIM   512
#define EPSF    1e-8f
#define LAMBDA_REGF 1.0f
#define ETAF    0.1f
#define KC      64    // K-panel depth staged in LDS
#define NB      64    // N-width per block (4 tiles of 16 per wave)

typedef float v2f __attribute__((ext_vector_type(2)));
typedef float v8f __attribute__((ext_vector_type(8)));

// ---------------------------------------------------------------- utilities
__global__ void zero_kernel(float* __restrict__ p, long n) {
  long i = (long)blockIdx.x * blockDim.x + threadIdx.x;
  long stride = (long)gridDim.x * blockDim.x;
  for (; i < n; i += stride) p[i] = 0.0f;
}

// col_sums[j] = sum of C_nonzero over edges with col==j
__global__ void colsum_kernel(const float* __restrict__ cnz,
                              const int* __restrict__ col,
                              float* __restrict__ colsum, int nnz) {
  int e = blockIdx.x * blockDim.x + threadIdx.x;
  if (e < nnz) atomicAdd(&colsum[col[e]], cnz[e]);
}

// vals = cnz / (colsum[col] + eps); scatter into dense C; reduce reg = sum(vals^2)
__global__ void scatter_kernel(const float* __restrict__ cnz,
                               const int* __restrict__ row,
                               const int* __restrict__ col,
                               const float* __restrict__ colsum,
                               float* __restrict__ C,
                               float* __restrict__ vals,
                               float* __restrict__ accum, int nnz) {
  int e = blockIdx.x * blockDim.x + threadIdx.x;
  float v = 0.0f;
  if (e < nnz) {
    int r = row[e], c = col[e];
    v = cnz[e] / (colsum[c] + EPSF);
    vals[e] = v;
    C[(long)r * N_NODES + c] = v;   // (row,col) pairs are unique -> plain store
  }
  __shared__ float red[256];
  red[threadIdx.x] = v * v;
  __syncthreads();
  for (int s = 128; s > 0; s >>= 1) {
    if (threadIdx.x < s) red[threadIdx.x] += red[threadIdx.x + s];
    __syncthreads();
  }
  if (threadIdx.x == 0) atomicAdd(&accum[1], red[0]);
}

__device__ __forceinline__ int lower_bound_i(const int* __restrict__ a, int n, int key) {
  int lo = 0, hi = n;
  while (lo < hi) { int mid = (lo + hi) >> 1; if (a[mid] < key) lo = mid + 1; else hi = mid; }
  return lo;
}

// lap1 = sum_e deg[row_e]*vals_e^2 ; lap2 = sum_{edges (i,k)} <C_i, C_k>
// edge list is CSR-sorted (row asc, col asc) -> sorted-list intersection.
__global__ void lap_kernel(const int* __restrict__ row,
                           const int* __restrict__ col,
                           const float* __restrict__ vals,
                           const float* __restrict__ deg,
                           float* __restrict__ accum, int nnz) {
  int e = blockIdx.x * blockDim.x + threadIdx.x;
  float l1 = 0.0f, l2 = 0.0f;
  if (e < nnz) {
    int i = row[e], k = col[e];
    float v = vals[e];
    l1 = deg[i] * v * v;
    int sa = lower_bound_i(row, nnz, i),  ea = lower_bound_i(row, nnz, i + 1);
    int sb = lower_bound_i(row, nnz, k),  eb = lower_bound_i(row, nnz, k + 1);
    int pa = sa, pb = sb;
    float dot = 0.0f;
    while (pa < ea && pb < eb) {
      int ca = col[pa], cb = col[pb];
      if (ca == cb)      { dot += vals[pa] * vals[pb]; ++pa; ++pb; }
      else if (ca < cb)  { ++pa; }
      else               { ++pb; }
    }
    l2 = dot;
  }
  __shared__ float r1[256], r2[256];
  r1[threadIdx.x] = l1; r2[threadIdx.x] = l2;
  __syncthreads();
  for (int s = 128; s > 0; s >>= 1) {
    if (threadIdx.x < s) { r1[threadIdx.x] += r1[threadIdx.x + s]; r2[threadIdx.x] += r2[threadIdx.x + s]; }
    __syncthreads();
  }
  if (threadIdx.x == 0) { atomicAdd(&accum[2], r1[0]); atomicAdd(&accum[3], r2[0]); }
}

// ------------------------------------------------ WMMA GEMM + fused recon loss
// R = Z^T @ C  (M=512 x N=4096, K=4096), f32 WMMA 16x16x4.
// Wave computes a 16x64 strip (4 accumulators, 4x A-fragment reuse).
// 8 waves/WG share a KCxNB C-panel, double-buffered in LDS and filled with
// CDNA5 async memory->LDS copies (ASYNCcnt).
__global__ void __launch_bounds__(256)
recon_wmma_kernel(const float* __restrict__ Z, const float* __restrict__ C,
                  float* __restrict__ recon_accum) {
  __shared__ float bsh[2][KC * NB];          // 2 x 16 KB

  const int tid   = threadIdx.x;
  const int lane  = tid & 31;
  const int wave  = tid >> 5;
  const int n0    = blockIdx.x * NB;               // 64 n-blocks
  const int d0    = (blockIdx.y * 8 + wave) * 16;  // 32 d-tiles
  const int halfk = lane >> 4;                     // 0: lanes 0-15, 1: lanes 16-31
  const int lidx  = lane & 15;

  v8f acc[4];
#pragma unroll
  for (int j = 0; j < 4; ++j) acc[j] = v8f{0.f,0.f,0.f,0.f,0.f,0.f,0.f,0.f};

  // async stage of C[k0:k0+KC, n0:n0+NB] into LDS buffer `b` (b128 granules)
  auto stage_async = [&](int b, int k0) {
    unsigned ldsbase = (unsigned)(uintptr_t)(&bsh[b][0]);
#pragma unroll
    for (int i = 0; i < (KC * NB / 4) / 256; ++i) {
      int idx = tid + i * 256;            // float4 index in [0, KC*NB/4)
      int kk  = idx >> 4;                 // KC rows, NB/4 = 16 float4 per row
      int nn  = (idx & 15) << 2;
      const float* gp = C + (long)(k0 + kk) * N_NODES + n0 + nn;
      unsigned lp = ldsbase + (unsigned)((kk * NB + nn) * 4);
      asm volatile("global_load_async_to_lds_b128 %0, %1, off"
                   :: "v"(lp), "v"(gp) : "memory");
    }
  };

  stage_async(0, 0);
  asm volatile("s_wait_asynccnt 0x0" ::: "memory");
  __syncthreads();

  int buf = 0;
  for (int k0 = 0; k0 < N_NODES; k0 += KC) {
    if (k0 + KC < N_NODES) stage_async(buf ^ 1, k0 + KC);   // overlap next panel

    const float* __restrict__ cur = &bsh[buf][0];
#pragma unroll 4
    for (int kk = 0; kk < KC; kk += 4) {
      const int krow = kk + 2 * halfk;
      // A fragment: A[m,k'] = Zt[d0+m, k0+krow+{0,1}] = Z[(k0+krow+{0,1})*D + d0+m]
      const float* zp = Z + (long)(k0 + krow) * D_DIM + d0 + lidx;
      v2f a;
      a.x = zp[0];
      a.y = zp[D_DIM];
#pragma unroll
      for (int j = 0; j < 4; ++j) {
        v2f b;
        b.x = cur[krow * NB + j * 16 + lidx];
        b.y = cur[(krow + 1) * NB + j * 16 + lidx];
        acc[j] = __builtin_amdgcn_wmma_f32_16x16x4_f32(false, a, false, b,
                                                       (short)0, acc[j], false, false);
      }
    }

    asm volatile("s_wait_asynccnt 0x0" ::: "memory");  // next panel landed
    __syncthreads();                                   // all waves done with cur
    buf ^= 1;
  }

  // epilogue: diff = acc - Zt[d,n]; accumulate sum of squares
  float s = 0.0f;
#pragma unroll
  for (int j = 0; j < 4; ++j) {
    int n = n0 + j * 16 + lidx;
    const float* zt = Z + (long)n * D_DIM + d0;   // Zt[d,n] = Z[n,d]
#pragma unroll
    for (int r = 0; r < 8; ++r) {
      float diff = acc[j][r] - zt[r + 8 * halfk]; // C/D layout: lanes 16-31 -> M+8
      s += diff * diff;
    }
  }
  for (int off = 16; off > 0; off >>= 1) s += __shfl_down(s, off, 32);
  if (lane == 0) atomicAdd(recon_accum, s);
}

__global__ void finalize_kernel(const float* __restrict__ accum,
                                float* __restrict__ out_scalars) {
  if (threadIdx.x == 0) {
    out_scalars[0] = accum[0] / (float)((long)D_DIM * N_NODES);  // recon_loss (mean)
    out_scalars[1] = LAMBDA_REGF * accum[1];                     // reg_loss
    out_scalars[2] = ETAF * (accum[2] - accum[3]);               // lap_loss
  }
}

// ---------------------------------------------------------------- launcher
extern "C" void kernel_launch(void* const* d_in, const int* in_sizes, int n_in,
                              void* d_out, int out_size, void* d_ws, size_t ws_size,
                              hipStream_t stream) {
  const float* Z    = (const float*)d_in[0];
  const float* cnz  = (const float*)d_in[1];
  const int*   row  = (const int*)d_in[2];
  const int*   col  = (const int*)d_in[3];
  const float* deg  = (const float*)d_in[4];
  const int    nnz  = in_sizes[1];

  float* C       = (float*)d_out;                       // N*N dense C
  float* scalars = C + (long)N_NODES * N_NODES;         // [recon, reg, lap]

  // workspace: col_sums | vals | accum[4] (recon, reg, lap1, lap2)
  float* colsum = (float*)d_ws;
  float* vals   = colsum + N_NODES;
  int nnzpad    = (nnz + 15) & ~15;
  float* accum  = vals + nnzpad;

  zero_kernel<<<4096, 256, 0, stream>>>(C, (long)N_NODES * N_NODES);
  zero_kernel<<<64, 256, 0, stream>>>(colsum, (long)(N_NODES + nnzpad + 16));

  int eb = (nnz + 255) / 256;
  colsum_kernel<<<eb, 256, 0, stream>>>(cnz, col, colsum, nnz);
  scatter_kernel<<<eb, 256, 0, stream>>>(cnz, row, col, colsum, C, vals, accum, nnz);
  lap_kernel<<<eb, 256, 0, stream>>>(row, col, vals, deg, accum, nnz);

  recon_wmma_kernel<<<dim3(N_NODES / NB, 4), 256, 0, stream>>>(Z, C, accum);

  finalize_kernel<<<1, 64, 0, stream>>>(accum, scalars);
}